// _PointnetSAModuleBase_88433376624924
// MI455X (gfx1250) — compile-verified
//
#include <hip/hip_runtime.h>

// PointNet++ SA module for MI455X (gfx1250), fp32 WMMA path.
// All matmul FLOPs via V_WMMA_F32_16X16X4_F32 (wave32, 16x16 tiles).
// Activations live in LDS in a "B-fragment-ready" layout so each lane's two
// K-values per WMMA are one ds_load_b64 at an immediate offset.

typedef float v2f __attribute__((ext_vector_type(2)));
typedef float v4f __attribute__((ext_vector_type(4)));
typedef float v8f __attribute__((ext_vector_type(8)));

#define NCOL   32          // columns per workgroup = S (one p-group)
#define NWAVES 8

// LDS layout (dword units): element (ch, col) at ((ch/4)*NCOL + col)*4 + (ch%4).
// Per-k-step (4 K values) stride = NCOL*4 dwords = 512 bytes.
__device__ __forceinline__ int lds_off(int ch, int col) {
    return ((ch >> 2) * NCOL + col) * 4 + (ch & 3);
}

// ---- single-tile layer (used for 64-out layers: 8 tiles / 8 waves) ---------
template <bool GATE, bool RELU>
__device__ __forceinline__ void layer_1tile(
    const float* __restrict__ W, const float* __restrict__ bias,
    const float* __restrict__ src, float* __restrict__ dst,
    const float* __restrict__ gate_src,
    int outBase, int colBase, int Kpad, int Kreal, int lane)
{
    const int row = lane & 15;
    const int kh  = lane >> 4;                 // K-half of this lane
    const int m0  = outBase + kh * 8;          // first output row in C VGPR 0
    v4f bl0 = *(const v4f*)(bias + m0);
    v4f bl1 = *(const v4f*)(bias + m0 + 4);
    v8f c = { bl0.x, bl0.y, bl0.z, bl0.w, bl1.x, bl1.y, bl1.z, bl1.w };

    const int col = colBase + row;
    const float* bsrc = src + (col << 2) + (kh << 1);              // + k-step*128
    const float* wrow = W + (outBase + row) * Kreal + (kh << 1);   // + k

    for (int k = 0; k < Kpad; k += 4) {
        const int kk = k + (kh << 1);
        v2f a;
        if (kk < Kreal) a = *(const v2f*)(wrow + k);   // 8B-aligned (K even)
        else            { a.x = 0.0f; a.y = 0.0f; }    // zero-padded K tail
        v2f bf = *(const v2f*)(bsrc + (k >> 2) * (NCOL * 4));
        c = __builtin_amdgcn_wmma_f32_16x16x4_f32(
                false, a, false, bf, (short)0, c, false, false);
    }

    v4f lo = { c[0], c[1], c[2], c[3] };
    v4f hi = { c[4], c[5], c[6], c[7] };
    if (GATE) {
        v4f g0 = *(const v4f*)(gate_src + (( m0      >> 2) * NCOL + col) * 4);
        v4f g1 = *(const v4f*)(gate_src + (((m0 + 4) >> 2) * NCOL + col) * 4);
        lo = g0 * (1.0f + lo);
        hi = g1 * (1.0f + hi);
    }
    if (RELU) {
#pragma unroll
        for (int j = 0; j < 4; ++j) {
            lo[j] = fmaxf(lo[j], 0.0f);
            hi[j] = fmaxf(hi[j], 0.0f);
        }
    }
    *(v4f*)(dst + (( m0      >> 2) * NCOL + col) * 4) = lo;
    *(v4f*)(dst + (((m0 + 4) >> 2) * NCOL + col) * 4) = hi;
}

// ---- paired-tile layer: one output tile row, both column tiles -------------
// Shares the A-fragment (weights) between two independent accumulator chains.
__device__ __forceinline__ void layer_2tile_relu(
    const float* __restrict__ W, const float* __restrict__ bias,
    const float* __restrict__ src, float* __restrict__ dst,
    int outBase, int K, int lane)
{
    const int row = lane & 15;
    const int kh  = lane >> 4;
    const int m0  = outBase + kh * 8;
    v4f bl0 = *(const v4f*)(bias + m0);
    v4f bl1 = *(const v4f*)(bias + m0 + 4);
    v8f c0 = { bl0.x, bl0.y, bl0.z, bl0.w, bl1.x, bl1.y, bl1.z, bl1.w };
    v8f c1 = c0;

    const float* bs0 = src + ( row       << 2) + (kh << 1);
    const float* bs1 = src + ((16 + row) << 2) + (kh << 1);
    const float* wrow = W + (outBase + row) * K + (kh << 1);

    for (int k = 0; k < K; k += 4) {
        v2f a  = *(const v2f*)(wrow + k);
        v2f f0 = *(const v2f*)(bs0 + (k >> 2) * (NCOL * 4));
        v2f f1 = *(const v2f*)(bs1 + (k >> 2) * (NCOL * 4));
        c0 = __builtin_amdgcn_wmma_f32_16x16x4_f32(
                 false, a, false, f0, (short)0, c0, false, false);
        c1 = __builtin_amdgcn_wmma_f32_16x16x4_f32(
                 false, a, false, f1, (short)0, c1, false, false);
    }

#pragma unroll
    for (int half = 0; half < 2; ++half) {
        const v8f& c = half ? c1 : c0;
        const int col = (half ? 16 : 0) + row;
        v4f lo = { fmaxf(c[0], 0.0f), fmaxf(c[1], 0.0f),
                   fmaxf(c[2], 0.0f), fmaxf(c[3], 0.0f) };
        v4f hi = { fmaxf(c[4], 0.0f), fmaxf(c[5], 0.0f),
                   fmaxf(c[6], 0.0f), fmaxf(c[7], 0.0f) };
        *(v4f*)(dst + (( m0      >> 2) * NCOL + col) * 4) = lo;
        *(v4f*)(dst + (((m0 + 4) >> 2) * NCOL + col) * 4) = hi;
    }
}

__global__ __launch_bounds__(256) void pointnet_sa_kernel(
    const float* __restrict__ xyz,      // (B, N, 3)
    const float* __restrict__ features, // (B, C, N)
    const float* __restrict__ new_xyz,  // (B, P, 3)
    const int*   __restrict__ idx,      // (B, P, S)
    const float* __restrict__ Wg, const float* __restrict__ bg,   // 64x106
    const float* __restrict__ W1, const float* __restrict__ b1,   // 64x64
    const float* __restrict__ W2, const float* __restrict__ b2,   // 128x64
    const float* __restrict__ W3, const float* __restrict__ b3,   // 256x128
    float* __restrict__ out_pooled)     // (B, 256, P)
{
    __shared__ float bufA[256 * NCOL];   // nf(108ch) -> l1(64) -> l3(256): 32KB
    __shared__ float bufB[128 * NCOL];   // gated x(64) -> l2(128): 16KB
    __shared__ int   s_idx[NCOL];
    __shared__ float s_nxyz[3];

    const int N_ = 16384, C_ = 103, P_ = 2048, S_ = 32;
    const int blk  = blockIdx.x;
    const int b    = blk / P_;
    const int p    = blk % P_;
    const int t    = threadIdx.x;
    const int wave = t >> 5;
    const int lane = t & 31;

    if (t < S_) s_idx[t]  = idx[(b * P_ + p) * S_ + t];
    if (t < 3)  s_nxyz[t] = new_xyz[(b * P_ + p) * 3 + t];
    __syncthreads();

    // Gather nf: ch 0-2 = relative xyz, ch 3-105 = features, ch 106-107 = 0.
    for (int e = t; e < 108 * NCOL; e += 256) {
        const int ch  = e >> 5;
        const int col = e & 31;
        float v = 0.0f;
        if (ch < 3)        v = xyz[(b * N_ + s_idx[col]) * 3 + ch] - s_nxyz[ch];
        else if (ch < 106) v = features[(b * C_ + (ch - 3)) * N_ + s_idx[col]];
        bufA[lds_off(ch, col)] = v;
    }
    __syncthreads();

    // Gating layer: w = Wg*nf + bg ; x = nf[:64] * (1 + w).  8 tiles / 8 waves.
    layer_1tile<true, false>(Wg, bg, bufA, bufB, bufA,
                             (wave >> 1) << 4, (wave & 1) << 4, 108, 106, lane);
    __syncthreads();
    // Layer 1: 64x64.  8 tiles / 8 waves.
    layer_1tile<false, true>(W1, b1, bufB, bufA, nullptr,
                             (wave >> 1) << 4, (wave & 1) << 4, 64, 64, lane);
    __syncthreads();
    // Layer 2: 128x64.  8 output tiles, each wave does both column tiles.
    layer_2tile_relu(W2, b2, bufA, bufB, wave << 4, 64, lane);
    __syncthreads();
    // Layer 3: 256x128.  16 output tiles, 2 per wave, both column tiles each.
    layer_2tile_relu(W3, b3, bufB, bufA, wave << 4, 128, lane);
    layer_2tile_relu(W3, b3, bufB, bufA, (wave + 8) << 4, 128, lane);
    __syncthreads();

    // Max-pool over S=32 columns; one thread per output channel, staggered
    // column order to spread LDS banks (worst case 2-way conflict).
    {
        const int ch = t;  // 0..255
        float m = -3.402823466e+38f;
#pragma unroll
        for (int s = 0; s < S_; ++s) {
            const int col = (s + t) & 31;
            m = fmaxf(m, bufA[lds_off(ch, col)]);
        }
        out_pooled[(b * 256 + ch) * P_ + p] = m;
    }
}

__global__ void copy_new_xyz_kernel(const float* __restrict__ src,
                                    float* __restrict__ dst, int n)
{
    const int i = blockIdx.x * blockDim.x + threadIdx.x;
    if (i < n) dst[i] = src[i];
}

extern "C" void kernel_launch(void* const* d_in, const int* in_sizes, int n_in,
                              void* d_out, int out_size, void* d_ws, size_t ws_size,
                              hipStream_t stream)
{
    (void)in_sizes; (void)n_in; (void)out_size; (void)d_ws; (void)ws_size;
    const float* xyz      = (const float*)d_in[0];
    const float* features = (const float*)d_in[1];
    const float* new_xyz  = (const float*)d_in[2];
    const int*   idx      = (const int*)  d_in[3];
    const float* Wg = (const float*)d_in[4];
    const float* bg = (const float*)d_in[5];
    const float* W1 = (const float*)d_in[6];
    const float* b1 = (const float*)d_in[7];
    const float* W2 = (const float*)d_in[8];
    const float* b2 = (const float*)d_in[9];
    const float* W3 = (const float*)d_in[10];
    const float* b3 = (const float*)d_in[11];
    float* out = (float*)d_out;

    const int B = 4, P = 2048;
    const int nxyz_n = B * P * 3;   // 24576 floats: first output (new_xyz)

    copy_new_xyz_kernel<<<(nxyz_n + 255) / 256, 256, 0, stream>>>(new_xyz, out, nxyz_n);

    pointnet_sa_kernel<<<B * P, 256, 0, stream>>>(
        xyz, features, new_xyz, idx,
        Wg, bg, W1, b1, W2, b2, W3, b3,
        out + nxyz_n);
}